// GCNModel_40398462386753
// MI455X (gfx1250) — compile-verified
//
#include <hip/hip_runtime.h>
#include <math.h>

typedef __attribute__((ext_vector_type(2))) float v2f;
typedef __attribute__((ext_vector_type(8))) float v8f;

#define N_NODES 50000
#define N_EDGES 800000

// =====================================================================
// GEMM: C[M x 256] = A[M x K] @ W[K x 256]   (f32, exact, WMMA 16x16x4)
// block = 128 threads (4 waves). blockIdx.x = 16-row band (3125 bands).
// wave w computes cols [64w, 64w+64) as 4 WMMA tiles sharing one A-frag.
// A-frag layout (ISA 7.12.2, 32-bit A 16x4): lane L -> M = L&15,
//   VGPR j / half-wave g -> K = 2*g + j  => per-lane contiguous b64 load.
// C/D layout: lane L -> N = L&15, VGPR j -> M = 8*(L>>4) + j.
// =====================================================================
__global__ __launch_bounds__(128)
void gemm_n256_wmma(const float* __restrict__ A, const float* __restrict__ W,
                    float* __restrict__ C, int K) {
    const int wave = threadIdx.x >> 5;
    const int lane = threadIdx.x & 31;
    const int g    = lane >> 4;     // half-wave select
    const int h    = lane & 15;
    const int row0 = blockIdx.x * 16;
    const int nb   = wave * 64;

    v8f acc0 = {}, acc1 = {}, acc2 = {}, acc3 = {};

    const float* aptr = A + (size_t)(row0 + h) * K + 2 * g;

    for (int k = 0; k < K; k += 4) {
        v2f afrag = *(const v2f*)(aptr + k);                 // b64 load
        const float* w0 = W + (size_t)(k + 2 * g) * 256;     // K row
        const float* w1 = w0 + 256;                          // K row +1

        v2f b0; b0.x = w0[nb + h];      b0.y = w1[nb + h];
        acc0 = __builtin_amdgcn_wmma_f32_16x16x4_f32(false, afrag, false, b0,
                                                     (short)0, acc0, false, false);
        v2f b1; b1.x = w0[nb + 16 + h]; b1.y = w1[nb + 16 + h];
        acc1 = __builtin_amdgcn_wmma_f32_16x16x4_f32(false, afrag, false, b1,
                                                     (short)0, acc1, false, false);
        v2f b2; b2.x = w0[nb + 32 + h]; b2.y = w1[nb + 32 + h];
        acc2 = __builtin_amdgcn_wmma_f32_16x16x4_f32(false, afrag, false, b2,
                                                     (short)0, acc2, false, false);
        v2f b3; b3.x = w0[nb + 48 + h]; b3.y = w1[nb + 48 + h];
        acc3 = __builtin_amdgcn_wmma_f32_16x16x4_f32(false, afrag, false, b3,
                                                     (short)0, acc3, false, false);
    }

#pragma unroll
    for (int j = 0; j < 8; ++j) {
        const int m = row0 + g * 8 + j;
        float* crow = C + (size_t)m * 256 + nb + h;
        crow[0]  = acc0[j];
        crow[16] = acc1[j];
        crow[32] = acc2[j];
        crow[48] = acc3[j];
    }
}

// =====================================================================
// GEMM: C[M x 40] = A[M x 256] @ W[256 x 40]  (3 masked 16-col tiles)
// block = 128 threads (4 waves); wave w -> 16-row band blockIdx.x*4+w.
// Early-exit is wave-uniform so EXEC stays all-ones for WMMA.
// =====================================================================
__global__ __launch_bounds__(128)
void gemm_n40_wmma(const float* __restrict__ A, const float* __restrict__ W,
                   float* __restrict__ C) {
    const int wave = threadIdx.x >> 5;
    const int lane = threadIdx.x & 31;
    const int g    = lane >> 4;
    const int h    = lane & 15;
    const int band = blockIdx.x * 4 + wave;
    if (band >= (N_NODES / 16)) return;          // wave-uniform
    const int row0 = band * 16;
    const int K = 256;

    v8f acc[3] = {{}, {}, {}};
    const float* aptr = A + (size_t)(row0 + h) * K + 2 * g;

    for (int k = 0; k < K; k += 4) {
        v2f afrag = *(const v2f*)(aptr + k);
        const float* w0 = W + (size_t)(k + 2 * g) * 40;
        const float* w1 = w0 + 40;
#pragma unroll
        for (int t = 0; t < 3; ++t) {
            const int n = t * 16 + h;
            v2f b;
            b.x = (n < 40) ? w0[n] : 0.0f;
            b.y = (n < 40) ? w1[n] : 0.0f;
            acc[t] = __builtin_amdgcn_wmma_f32_16x16x4_f32(false, afrag, false, b,
                                                           (short)0, acc[t], false, false);
        }
    }

#pragma unroll
    for (int j = 0; j < 8; ++j) {
        const int m = row0 + g * 8 + j;
#pragma unroll
        for (int t = 0; t < 3; ++t) {
            const int n = t * 16 + h;
            if (n < 40) C[(size_t)m * 40 + n] = acc[t][j];
        }
    }
}

// =====================================================================
// SpMM scatter: Acc[row[e]][:] += vals[e] * S[col[e]][:]
// one thread = one edge x 4 channels; HW f32 atomics resolved in L2.
// =====================================================================
__global__ __launch_bounds__(256)
void scatter_edges(const int* __restrict__ erow, const int* __restrict__ ecol,
                   const float* __restrict__ evals,
                   const float* __restrict__ S, float* __restrict__ Acc,
                   int C, int groups) {
    const unsigned gid = blockIdx.x * 256u + threadIdx.x;
    const unsigned total = (unsigned)N_EDGES * (unsigned)groups;
    if (gid >= total) return;
    const int e  = gid / groups;
    const int cg = gid % groups;
    const int r = erow[e];
    const int c = ecol[e];
    const float v = evals[e];
    const float4 s = *(const float4*)(S + (size_t)c * C + cg * 4);
    float* dst = Acc + (size_t)r * C + cg * 4;
    unsafeAtomicAdd(dst + 0, v * s.x);
    unsafeAtomicAdd(dst + 1, v * s.y);
    unsafeAtomicAdd(dst + 2, v * s.z);
    unsafeAtomicAdd(dst + 3, v * s.w);
}

// =====================================================================
// H = max(H + b, 0), in place
// =====================================================================
__global__ __launch_bounds__(256)
void bias_relu(float* __restrict__ H, const float* __restrict__ b,
               int C, unsigned total) {
    const unsigned gid = blockIdx.x * 256u + threadIdx.x;
    if (gid >= total) return;
    const int c = gid % C;
    const float x = H[gid] + b[c];
    H[gid] = x > 0.0f ? x : 0.0f;
}

// =====================================================================
// out[i,:] = log_softmax(Acc[i,:] + b2), 40 classes, one thread per node
// =====================================================================
__global__ __launch_bounds__(256)
void bias_logsoftmax40(const float* __restrict__ Acc, const float* __restrict__ b,
                       float* __restrict__ out) {
    const int i = blockIdx.x * 256 + threadIdx.x;
    if (i >= N_NODES) return;
    float z[40];
    float mx = -INFINITY;
#pragma unroll
    for (int j = 0; j < 40; ++j) {
        z[j] = Acc[(size_t)i * 40 + j] + b[j];
        mx = fmaxf(mx, z[j]);
    }
    float s = 0.0f;
#pragma unroll
    for (int j = 0; j < 40; ++j) s += expf(z[j] - mx);
    const float l = mx + logf(s);
#pragma unroll
    for (int j = 0; j < 40; ++j) out[(size_t)i * 40 + j] = z[j] - l;
}

extern "C" void kernel_launch(void* const* d_in, const int* in_sizes, int n_in,
                              void* d_out, int out_size, void* d_ws, size_t ws_size,
                              hipStream_t stream) {
    const float* x    = (const float*)d_in[0];
    const float* ev   = (const float*)d_in[1];
    const float* W0   = (const float*)d_in[2];
    const float* b0   = (const float*)d_in[3];
    const float* W1   = (const float*)d_in[4];
    const float* b1   = (const float*)d_in[5];
    const float* W2   = (const float*)d_in[6];
    const float* b2   = (const float*)d_in[7];
    const int*   erow = (const int*)d_in[8];
    const int*   ecol = (const int*)d_in[9];
    float* out = (float*)d_out;

    const size_t BUF = (size_t)N_NODES * 256;   // floats (51.2 MB)
    float* buf0 = (float*)d_ws;                 // GEMM outputs (support)
    float* buf1 = buf0 + BUF;                   // SpMM accumulators / H

    const int GEMM_BLOCKS = N_NODES / 16;       // 3125
    const int SCAT_BLOCKS = (N_EDGES * 64) / 256;          // 200000
    const int RELU_BLOCKS = (N_NODES * 256) / 256;         // 50000
    const int G40_BLOCKS  = (GEMM_BLOCKS + 3) / 4;         // 782
    const int SCAT2_BLOCKS = (N_EDGES * 10 + 255) / 256;   // 31250
    const int LSM_BLOCKS  = (N_NODES + 255) / 256;         // 196

    // ---- Layer 0: S0 = x @ W0 ; A0 = adj @ S0 ; H0 = relu(A0 + b0) ----
    gemm_n256_wmma<<<GEMM_BLOCKS, 128, 0, stream>>>(x, W0, buf0, 512);
    hipMemsetAsync(buf1, 0, BUF * sizeof(float), stream);
    scatter_edges<<<SCAT_BLOCKS, 256, 0, stream>>>(erow, ecol, ev, buf0, buf1, 256, 64);
    bias_relu<<<RELU_BLOCKS, 256, 0, stream>>>(buf1, b0, 256, (unsigned)(N_NODES * 256));

    // ---- Layer 1: S1 = H0 @ W1 ; A1 = adj @ S1 ; H1 = relu(A1 + b1) ----
    gemm_n256_wmma<<<GEMM_BLOCKS, 128, 0, stream>>>(buf1, W1, buf0, 256);
    hipMemsetAsync(buf1, 0, BUF * sizeof(float), stream);   // H0 dead after gemm
    scatter_edges<<<SCAT_BLOCKS, 256, 0, stream>>>(erow, ecol, ev, buf0, buf1, 256, 64);
    bias_relu<<<RELU_BLOCKS, 256, 0, stream>>>(buf1, b1, 256, (unsigned)(N_NODES * 256));

    // ---- Layer 2: S2 = H1 @ W2 ; A2 = adj @ S2 ; out = log_softmax(A2+b2) ----
    gemm_n40_wmma<<<G40_BLOCKS, 128, 0, stream>>>(buf1, W2, buf0);
    hipMemsetAsync(buf1, 0, (size_t)N_NODES * 40 * sizeof(float), stream); // H1 dead
    scatter_edges<<<SCAT2_BLOCKS, 256, 0, stream>>>(erow, ecol, ev, buf0, buf1, 40, 10);
    bias_logsoftmax40<<<LSM_BLOCKS, 256, 0, stream>>>(buf1, b2, out);
}